// FastRPModel_36498632081942
// MI455X (gfx1250) — compile-verified
//
#include <hip/hip_runtime.h>
#include <hip/hip_bf16.h>

typedef __attribute__((ext_vector_type(2))) float v2f;
typedef __attribute__((ext_vector_type(8))) float v8f;

// softmax over K=3 within each of P=4 paths; planes flattened j = p*3 + k
__device__ __forceinline__ void softmax12(const float* __restrict__ fw, float* w12) {
#pragma unroll
  for (int p = 0; p < 4; ++p) {
    float f0 = fw[p * 3 + 0], f1 = fw[p * 3 + 1], f2 = fw[p * 3 + 2];
    float mx = fmaxf(f0, fmaxf(f1, f2));
    float e0 = __expf(f0 - mx), e1 = __expf(f1 - mx), e2 = __expf(f2 - mx);
    float inv = 1.0f / (e0 + e1 + e2);
    w12[p * 3 + 0] = e0 * inv;
    w12[p * 3 + 1] = e1 * inv;
    w12[p * 3 + 2] = e2 * inv;
  }
}

// Stage 1: emb[m] = sum_{j=0..11} w_j * F[j*ND + m], via V_WMMA_F32_16X16X4_F32.
// A (16x4): all rows = weight chunk  -> lane: V0 = w[4s+2*half], V1 = w[4s+2*half+1]
// B (4x16): 16 consecutive elements  -> lane: V0 = F[plane 4s+2*half][m0+col], V1 = next plane
// Every row of D is identical => c[0] of any lane = out[m0 + col].
//
// Two adjacent 16-element tiles per iteration: lanes 0..15 store tile A, lanes
// 16..31 store tile B -> one unmasked coalesced 128B store, no EXEC branch.
// Base pointers are laundered through readfirstlane so they live in SGPRs and
// loads use the SADDR + 32-bit-voffset form (no per-load 64-bit VALU adds).
__global__ void fastrp_emb_wmma(const float* __restrict__ F, const float* __restrict__ fw,
                                float* __restrict__ emb, long long ND, int npairs) {
  const int lane = (int)(threadIdx.x & 31);
  const int half = lane >> 4;   // which 16-lane half of the wave
  const int col  = lane & 15;   // N column / element within tile

  const int wave   = __builtin_amdgcn_readfirstlane(
                       (int)((blockIdx.x * blockDim.x + threadIdx.x) >> 5));
  const int nwaves = (int)((gridDim.x * blockDim.x) >> 5);

  float w12[12];
  softmax12(fw, w12);

  // A operands: constant across all tiles (rows of A identical on purpose)
  v2f a0, a1, a2;
  a0.x = w12[0 + 2 * half]; a0.y = w12[1 + 2 * half];   // chunk 0: planes 0..3
  a1.x = w12[4 + 2 * half]; a1.y = w12[5 + 2 * half];   // chunk 1: planes 4..7
  a2.x = w12[8 + 2 * half]; a2.y = w12[9 + 2 * half];   // chunk 2: planes 8..11

  // loop-invariant per-lane element offsets; signed 32-bit (max ~176M, byte
  // offset ~704MB < 2^31) so the GVS sext(voffset) addressing form applies
  const int v00 = (int)((long long)(0 + 2 * half) * ND) + col;
  const int v01 = (int)((long long)(1 + 2 * half) * ND) + col;
  const int v10 = (int)((long long)(4 + 2 * half) * ND) + col;
  const int v11 = (int)((long long)(5 + 2 * half) * ND) + col;
  const int v20 = (int)((long long)(8 + 2 * half) * ND) + col;
  const int v21 = (int)((long long)(9 + 2 * half) * ND) + col;

  const size_t step = (size_t)nwaves * 32;
  const float* Fm = F + (size_t)wave * 32;    // scalar (SGPR) base, advances per iter
  float*       Em = emb + (size_t)wave * 32;  // scalar (SGPR) base, advances per iter

  for (int q = wave; q < npairs; q += nwaves, Fm += step, Em += step) {
    // streaming reads: non-temporal so they don't thrash L2 (emb must stay resident)
    v2f b0, b1, b2, d0, d1, d2;
    b0.x = __builtin_nontemporal_load(Fm + v00);
    b0.y = __builtin_nontemporal_load(Fm + v01);
    b1.x = __builtin_nontemporal_load(Fm + v10);
    b1.y = __builtin_nontemporal_load(Fm + v11);
    b2.x = __builtin_nontemporal_load(Fm + v20);
    b2.y = __builtin_nontemporal_load(Fm + v21);
    d0.x = __builtin_nontemporal_load(Fm + v00 + 16);   // tile B: same voffset regs,
    d0.y = __builtin_nontemporal_load(Fm + v01 + 16);   // instruction offset:64
    d1.x = __builtin_nontemporal_load(Fm + v10 + 16);
    d1.y = __builtin_nontemporal_load(Fm + v11 + 16);
    d2.x = __builtin_nontemporal_load(Fm + v20 + 16);
    d2.y = __builtin_nontemporal_load(Fm + v21 + 16);

    v8f cA = {}, cB = {};
    // interleave the two independent accumulator chains to hide D->C latency
    cA = __builtin_amdgcn_wmma_f32_16x16x4_f32(false, a0, false, b0, (short)0, cA, false, false);
    cB = __builtin_amdgcn_wmma_f32_16x16x4_f32(false, a0, false, d0, (short)0, cB, false, false);
    cA = __builtin_amdgcn_wmma_f32_16x16x4_f32(false, a1, false, b1, (short)0, cA, false, false);
    cB = __builtin_amdgcn_wmma_f32_16x16x4_f32(false, a1, false, d1, (short)0, cB, false, false);
    cA = __builtin_amdgcn_wmma_f32_16x16x4_f32(false, a2, false, b2, (short)0, cA, false, false);
    cB = __builtin_amdgcn_wmma_f32_16x16x4_f32(false, a2, false, d2, (short)0, cB, false, false);

    // lane L holds out[m0+col] of its tile in every D VGPR; lanes 0..15 cover
    // elements 0..15 (tile A), lanes 16..31 cover 16..31 (tile B).
    Em[lane] = (half == 0) ? cA[0] : cB[0];
  }
}

// Stage 2: one edge per 16-lane group; float4 per lane covers D=64.
__global__ void fastrp_edges(const float* __restrict__ emb, const float* __restrict__ ic,
                             const int* __restrict__ ii, const int* __restrict__ jj,
                             float* __restrict__ out, int E) {
  const int gid = blockIdx.x * blockDim.x + threadIdx.x;
  const int e = gid >> 4;
  const int l = gid & 15;
  if (e >= E) return;

  const size_t oi = (size_t)ii[e] * 64 + (size_t)(4 * l);
  const size_t oj = (size_t)jj[e] * 64 + (size_t)(4 * l);
  const float4 a = *(const float4*)(emb + oi);
  const float4 b = *(const float4*)(emb + oj);

  const float dx = a.x - b.x, dy = a.y - b.y, dz = a.z - b.z, dw = a.w - b.w;
  float part = dx * dx + dy * dy + dz * dz + dw * dw;
  part += __shfl_xor(part, 8, 16);
  part += __shfl_xor(part, 4, 16);
  part += __shfl_xor(part, 2, 16);
  part += __shfl_xor(part, 1, 16);

  if (l == 0) {
    const float logit = ic[0] - part * (1.0f / 64.0f);
    out[e] = 1.0f / (1.0f + __expf(-logit));
  }
}

// Fallback if d_ws is too small for the 64MB emb table: recompute rows per edge.
__global__ void fastrp_fused(const float* __restrict__ F, const float* __restrict__ fw,
                             const float* __restrict__ ic, const int* __restrict__ ii,
                             const int* __restrict__ jj, float* __restrict__ out,
                             long long ND, int E) {
  const int gid = blockIdx.x * blockDim.x + threadIdx.x;
  const int e = gid >> 4;
  const int l = gid & 15;
  if (e >= E) return;

  float w12[12];
  softmax12(fw, w12);

  const size_t oi = (size_t)ii[e] * 64 + (size_t)(4 * l);
  const size_t oj = (size_t)jj[e] * 64 + (size_t)(4 * l);

  float zix = 0.f, ziy = 0.f, ziz = 0.f, ziw = 0.f;
  float zjx = 0.f, zjy = 0.f, zjz = 0.f, zjw = 0.f;
#pragma unroll
  for (int p = 0; p < 12; ++p) {
    const float4 fi = *(const float4*)(F + (size_t)p * (size_t)ND + oi);
    const float4 fj = *(const float4*)(F + (size_t)p * (size_t)ND + oj);
    const float w = w12[p];
    zix += w * fi.x; ziy += w * fi.y; ziz += w * fi.z; ziw += w * fi.w;
    zjx += w * fj.x; zjy += w * fj.y; zjz += w * fj.z; zjw += w * fj.w;
  }
  const float dx = zix - zjx, dy = ziy - zjy, dz = ziz - zjz, dw = ziw - zjw;
  float part = dx * dx + dy * dy + dz * dz + dw * dw;
  part += __shfl_xor(part, 8, 16);
  part += __shfl_xor(part, 4, 16);
  part += __shfl_xor(part, 2, 16);
  part += __shfl_xor(part, 1, 16);

  if (l == 0) {
    const float logit = ic[0] - part * (1.0f / 64.0f);
    out[e] = 1.0f / (1.0f + __expf(-logit));
  }
}

extern "C" void kernel_launch(void* const* d_in, const int* in_sizes, int n_in,
                              void* d_out, int out_size, void* d_ws, size_t ws_size,
                              hipStream_t stream) {
  const float* F  = (const float*)d_in[0];   // [P,K,N,D] = 12 planes of ND floats
  const float* fw = (const float*)d_in[1];   // [P,K] = 12
  const float* ic = (const float*)d_in[2];   // scalar intercept
  const int*   ii = (const int*)d_in[3];     // [E]
  const int*   jj = (const int*)d_in[4];     // [E]
  float* out = (float*)d_out;

  const long long PK = in_sizes[1];                       // 12
  const long long ND = (long long)in_sizes[0] / PK;       // N*D = 16,000,000
  const int E = in_sizes[3];

  const int edge_threads = E * 16;
  const int edge_blocks = (edge_threads + 255) / 256;

  if (ws_size >= (size_t)ND * sizeof(float)) {
    float* emb = (float*)d_ws;
    const int npairs = (int)(ND / 32);                    // D=64 => ND % 32 == 0
    fastrp_emb_wmma<<<4096, 256, 0, stream>>>(F, fw, emb, ND, npairs);
    fastrp_edges<<<edge_blocks, 256, 0, stream>>>(emb, ic, ii, jj, out, E);
  } else {
    fastrp_fused<<<edge_blocks, 256, 0, stream>>>(F, fw, ic, ii, jj, out, ND, E);
  }
}